// BasicBlock_66365834658163
// MI455X (gfx1250) — compile-verified
//
#include <hip/hip_runtime.h>
#include <math.h>

#define N_NODES 100000
#define N_EDGES 640000
#define DIM 128

static_assert(N_NODES % 16 == 0, "node tiling assumes N %% 16 == 0");

typedef float v2f __attribute__((ext_vector_type(2)));
typedef float v8f __attribute__((ext_vector_type(8)));
typedef float f4  __attribute__((ext_vector_type(4)));   // POD float4 (AS-safe)

// Global (device memory) address space casts: force global_load/global_atomic
// instead of flat_* (flat ties up DScnt / the LDS path on CDNA5).
#define GAS __attribute__((address_space(1)))
typedef const GAS float* gcf;
typedef GAS float*       gf;
typedef const GAS f4*    gcf4;
typedef GAS f4*          gf4;
typedef const GAS int*   gci;

__device__ __forceinline__ void gatomic_add(float* p, float v) {
    __hip_atomic_fetch_add((GAS float*)p, v, __ATOMIC_RELAXED,
                           __HIP_MEMORY_SCOPE_AGENT);
}
__device__ __forceinline__ void gatomic_max(float* p, float v) {
    __hip_atomic_fetch_max((GAS float*)p, v, __ATOMIC_RELAXED,
                           __HIP_MEMORY_SCOPE_AGENT);
}

// ---------------------------------------------------------------------------
// Kernel 0: initialize workspace accumulators
// acc [N*128]=0 ; m [N*2]=-inf ; den [N*2]=0 ; deg [N]=1 ; bzero [128]=0
// ---------------------------------------------------------------------------
__global__ void init_ws_kernel(float* __restrict__ acc, float* __restrict__ m,
                               float* __restrict__ den, float* __restrict__ deg,
                               float* __restrict__ bzero) {
    int i = blockIdx.x * blockDim.x + threadIdx.x;
    if (i < N_NODES * DIM) ((gf)acc)[i] = 0.0f;
    if (i < 2 * N_NODES) { ((gf)m)[i] = -INFINITY; ((gf)den)[i] = 0.0f; }
    if (i < N_NODES) ((gf)deg)[i] = 1.0f;
    if (i < DIM) ((gf)bzero)[i] = 0.0f;
}

// ---------------------------------------------------------------------------
// One 16x16 output tile of Y = X*W + b via 32 x V_WMMA_F32_16X16X4_F32.
// A fragment (16x4 f32): lanes 0-15 -> K=k0,k0+1 ; lanes 16-31 -> K=k0+2,k0+3
// B fragment (4x16 f32): mirrored K halves, N = lane&15
// C/D (16x16 f32): vgpr r: lanes0-15 M=r, lanes16-31 M=r+8, N=lane&15
// ---------------------------------------------------------------------------
__device__ __forceinline__ void gemm_tile_16x16(
    const float* __restrict__ xs,   // LDS tile, 16 rows x stride 132
    gcf W, gcf bias, gf y,
    int row0, int nbase, int half, int l15) {
    v8f c = {0.f, 0.f, 0.f, 0.f, 0.f, 0.f, 0.f, 0.f};
    #pragma unroll 4
    for (int k0 = 0; k0 < DIM; k0 += 4) {
        int kk = k0 + 2 * half;
        v2f a, b;
        a.x = xs[l15 * 132 + kk];
        a.y = xs[l15 * 132 + kk + 1];
        b.x = W[kk * DIM + nbase + l15];
        b.y = W[(kk + 1) * DIM + nbase + l15];
        c = __builtin_amdgcn_wmma_f32_16x16x4_f32(
                false, a, false, b, (short)0, c, false, false);
    }
    float bv = bias[nbase + l15];
    #pragma unroll
    for (int r = 0; r < 8; ++r) {
        int gm = row0 + r + 8 * half;          // always < N (N % 16 == 0)
        y[(size_t)gm * DIM + nbase + l15] = c[r] + bv;
    }
}

// ---------------------------------------------------------------------------
// Kernel 1: fused 5-way projection GEMM. One block = 16 node rows, 8 waves;
// wave w owns output columns [16w, 16w+16) for all five weight matrices.
// ---------------------------------------------------------------------------
__global__ __launch_bounds__(256) void proj_gemm_kernel(
    const float* __restrict__ x,
    const float* __restrict__ W0, const float* __restrict__ b0, float* __restrict__ y0,
    const float* __restrict__ W1, const float* __restrict__ b1, float* __restrict__ y1,
    const float* __restrict__ W2, const float* __restrict__ b2, float* __restrict__ y2,
    const float* __restrict__ W3, const float* __restrict__ b3, float* __restrict__ y3,
    const float* __restrict__ W4, const float* __restrict__ b4, float* __restrict__ y4) {
    __shared__ float xs[16 * 132];   // 16x128 tile, stride 132 kills bank conflicts

    const int tid  = threadIdx.x;
    const int row0 = blockIdx.x * 16;

    // cooperative, coalesced load of the 16x128 x-tile (8 elems/thread)
    gcf xg = (gcf)x;
    #pragma unroll
    for (int j = 0; j < 8; ++j) {
        int idx = tid + j * 256;
        int r   = idx >> 7;
        int col = idx & 127;
        xs[r * 132 + col] = xg[(size_t)(row0 + r) * DIM + col];
    }
    __syncthreads();

    const int wave  = tid >> 5;
    const int lane  = tid & 31;
    const int half  = lane >> 4;   // 0 = lanes 0-15, 1 = lanes 16-31
    const int l15   = lane & 15;
    const int nbase = wave * 16;

    gemm_tile_16x16(xs, (gcf)W0, (gcf)b0, (gf)y0, row0, nbase, half, l15);
    gemm_tile_16x16(xs, (gcf)W1, (gcf)b1, (gf)y1, row0, nbase, half, l15);
    gemm_tile_16x16(xs, (gcf)W2, (gcf)b2, (gf)y2, row0, nbase, half, l15);
    gemm_tile_16x16(xs, (gcf)W3, (gcf)b3, (gf)y3, row0, nbase, half, l15);
    gemm_tile_16x16(xs, (gcf)W4, (gcf)b4, (gf)y4, row0, nbase, half, l15);
}

// ---------------------------------------------------------------------------
// Kernel 2: per-edge attention logits + segment max + degree count.
// One wave per edge: lane l loads f4 of q[dst] and k[src] at channel 4l.
// Lanes 0-15 reduce head 0, lanes 16-31 reduce head 1 (C = 64 ch/head).
// ---------------------------------------------------------------------------
__global__ __launch_bounds__(256) void edge_logits_kernel(
    const int* __restrict__ ei, const float* __restrict__ q,
    const float* __restrict__ k, float* __restrict__ logits,
    float* __restrict__ m, float* __restrict__ deg) {
    int wid  = (blockIdx.x * blockDim.x + threadIdx.x) >> 5;
    int lane = threadIdx.x & 31;
    if (wid >= N_EDGES) return;

    int src = ((gci)ei)[wid];
    int dst = ((gci)ei)[N_EDGES + wid];
    f4 qa = ((gcf4)(q + (size_t)dst * DIM))[lane];
    f4 ka = ((gcf4)(k + (size_t)src * DIM))[lane];
    float p = qa.x * ka.x + qa.y * ka.y + qa.z * ka.z + qa.w * ka.w;
    // reduce within each 16-lane half (one head each)
    p += __shfl_xor(p, 1, 32);
    p += __shfl_xor(p, 2, 32);
    p += __shfl_xor(p, 4, 32);
    p += __shfl_xor(p, 8, 32);
    if ((lane & 15) == 0) {
        int h = lane >> 4;
        float lg = p * 0.125f;                     // 1/sqrt(64)
        ((gf)logits)[(size_t)wid * 2 + h] = lg;
        gatomic_max(&m[(size_t)dst * 2 + h], lg);
        if (h == 0) gatomic_add(&deg[dst], 1.0f);  // in-degree (self-loop preset)
    }
}

// ---------------------------------------------------------------------------
// Kernel 3: exp(logit - segmax) and softmax denominator (segment sum).
// One thread per (edge, head). Overwrites logits with exp values.
// ---------------------------------------------------------------------------
__global__ void edge_exp_kernel(const int* __restrict__ ei,
                                float* __restrict__ logits,
                                const float* __restrict__ m,
                                float* __restrict__ den) {
    int t = blockIdx.x * blockDim.x + threadIdx.x;
    if (t >= 2 * N_EDGES) return;
    int e = t >> 1;
    int h = t & 1;
    int dst = ((gci)ei)[N_EDGES + e];
    float ex = __expf(((gcf)logits)[t] - ((gcf)m)[(size_t)dst * 2 + h]);
    ((gf)logits)[t] = ex;
    gatomic_add(&den[(size_t)dst * 2 + h], ex);
}

// ---------------------------------------------------------------------------
// Kernel 4: fused edge scatter. One wave per edge; lane l owns channels 4l..4l+3.
//   acc[dst] += alpha * v[src]  (attention)  +  d^-1/2_src d^-1/2_dst * xw[src]
// ---------------------------------------------------------------------------
__global__ __launch_bounds__(256) void edge_scatter_kernel(
    const int* __restrict__ ei, const float* __restrict__ logits,
    const float* __restrict__ den, const float* __restrict__ deg,
    const float* __restrict__ v, const float* __restrict__ xw,
    float* __restrict__ acc) {
    int wid  = (blockIdx.x * blockDim.x + threadIdx.x) >> 5;
    int lane = threadIdx.x & 31;
    if (wid >= N_EDGES) return;

    int src = ((gci)ei)[wid];
    int dst = ((gci)ei)[N_EDGES + wid];
    int h = lane >> 4;
    float ex    = ((gcf)logits)[(size_t)wid * 2 + h];
    float alpha = ex / (((gcf)den)[(size_t)dst * 2 + h] + 1e-16f);
    float gn    = rsqrtf(((gcf)deg)[src]) * rsqrtf(((gcf)deg)[dst]);

    f4 vv = ((gcf4)(v  + (size_t)src * DIM))[lane];
    f4 xx = ((gcf4)(xw + (size_t)src * DIM))[lane];
    float* a = acc + (size_t)dst * DIM + lane * 4;
    gatomic_add(a + 0, alpha * vv.x + gn * xx.x);
    gatomic_add(a + 1, alpha * vv.y + gn * xx.y);
    gatomic_add(a + 2, alpha * vv.z + gn * xx.z);
    gatomic_add(a + 3, alpha * vv.w + gn * xx.w);
}

// ---------------------------------------------------------------------------
// Kernel 5: residual + GCN self-loop + bias + LayerNorm + ReLU.
// One wave per node; full 32-lane shfl reduction for mean/var over 128 ch.
// ---------------------------------------------------------------------------
__global__ __launch_bounds__(256) void finalize_kernel(
    const float* __restrict__ x, const float* __restrict__ acc,
    const float* __restrict__ hsk, const float* __restrict__ xw,
    const float* __restrict__ deg, const float* __restrict__ bgcn,
    const float* __restrict__ gamma, const float* __restrict__ beta,
    float* __restrict__ out) {
    int wid  = (blockIdx.x * blockDim.x + threadIdx.x) >> 5;
    int lane = threadIdx.x & 31;
    if (wid >= N_NODES) return;

    float inv_deg = 1.0f / ((gcf)deg)[wid];   // dis*dis self-loop weight
    size_t base = (size_t)wid * DIM;

    f4 xv = ((gcf4)(x   + base))[lane];
    f4 av = ((gcf4)(acc + base))[lane];
    f4 hv = ((gcf4)(hsk + base))[lane];
    f4 wv = ((gcf4)(xw  + base))[lane];
    f4 bv = ((gcf4)(bgcn))[lane];

    float t[4];
    t[0] = xv.x + av.x + hv.x + inv_deg * wv.x + bv.x;
    t[1] = xv.y + av.y + hv.y + inv_deg * wv.y + bv.y;
    t[2] = xv.z + av.z + hv.z + inv_deg * wv.z + bv.z;
    t[3] = xv.w + av.w + hv.w + inv_deg * wv.w + bv.w;

    float s = 0.0f, sq = 0.0f;
    #pragma unroll
    for (int j = 0; j < 4; ++j) { s += t[j]; sq += t[j] * t[j]; }
    #pragma unroll
    for (int off = 1; off < 32; off <<= 1) {
        s  += __shfl_xor(s, off, 32);
        sq += __shfl_xor(sq, off, 32);
    }
    float mu   = s * (1.0f / 128.0f);
    float var  = sq * (1.0f / 128.0f) - mu * mu;
    float rstd = rsqrtf(var + 1e-5f);

    f4 gm = ((gcf4)(gamma))[lane];
    f4 bt = ((gcf4)(beta))[lane];
    f4 o;
    o.x = fmaxf((t[0] - mu) * rstd * gm.x + bt.x, 0.0f);
    o.y = fmaxf((t[1] - mu) * rstd * gm.y + bt.y, 0.0f);
    o.z = fmaxf((t[2] - mu) * rstd * gm.z + bt.z, 0.0f);
    o.w = fmaxf((t[3] - mu) * rstd * gm.w + bt.w, 0.0f);
    ((gf4)(out + base))[lane] = o;
}

// ---------------------------------------------------------------------------
// Host launcher
// ---------------------------------------------------------------------------
extern "C" void kernel_launch(void* const* d_in, const int* in_sizes, int n_in,
                              void* d_out, int out_size, void* d_ws, size_t ws_size,
                              hipStream_t stream) {
    const float* x     = (const float*)d_in[0];
    const int*   ei    = (const int*)d_in[1];
    const float* Wq    = (const float*)d_in[2];
    const float* bq    = (const float*)d_in[3];
    const float* Wk    = (const float*)d_in[4];
    const float* bk    = (const float*)d_in[5];
    const float* Wv    = (const float*)d_in[6];
    const float* bv    = (const float*)d_in[7];
    const float* Wsk   = (const float*)d_in[8];
    const float* bsk   = (const float*)d_in[9];
    const float* Wg    = (const float*)d_in[10];
    const float* bg    = (const float*)d_in[11];
    const float* gamma = (const float*)d_in[12];
    const float* beta  = (const float*)d_in[13];
    float* out = (float*)d_out;

    // ---- workspace layout (floats) ----
    const size_t ND = (size_t)N_NODES * DIM;
    float* ws  = (float*)d_ws;
    size_t off = 0;
    float* q      = ws + off; off += ND;
    float* kbuf   = ws + off; off += ND;
    float* vbuf   = ws + off; off += ND;
    float* hsk    = ws + off; off += ND;
    float* xw     = ws + off; off += ND;
    float* acc    = ws + off; off += ND;
    float* logits = ws + off; off += (size_t)N_EDGES * 2;
    float* m      = ws + off; off += (size_t)N_NODES * 2;
    float* den    = ws + off; off += (size_t)N_NODES * 2;
    float* deg    = ws + off; off += (size_t)N_NODES;
    float* bzero  = ws + off; off += DIM;
    if (ws_size < off * sizeof(float)) return;  // insufficient scratch

    // 0) init accumulators
    {
        int total = N_NODES * DIM;
        init_ws_kernel<<<(total + 255) / 256, 256, 0, stream>>>(acc, m, den, deg,
                                                                bzero);
    }
    // 1) fused projections: q,k,v (bias), skip (bias), xw (zero bias)
    {
        int blocks = N_NODES / 16;
        proj_gemm_kernel<<<blocks, 256, 0, stream>>>(
            x,
            Wq,  bq,    q,
            Wk,  bk,    kbuf,
            Wv,  bv,    vbuf,
            Wsk, bsk,   hsk,
            Wg,  bzero, xw);
    }
    // 2) logits + segment max + degree
    {
        int blocks = (N_EDGES + 7) / 8;   // 8 waves (edges) per 256-thread block
        edge_logits_kernel<<<blocks, 256, 0, stream>>>(ei, q, kbuf, logits, m, deg);
    }
    // 3) exp + denominator
    {
        int total = 2 * N_EDGES;
        edge_exp_kernel<<<(total + 255) / 256, 256, 0, stream>>>(ei, logits, m, den);
    }
    // 4) fused attention + GCN scatter
    {
        int blocks = (N_EDGES + 7) / 8;
        edge_scatter_kernel<<<blocks, 256, 0, stream>>>(ei, logits, den, deg,
                                                        vbuf, xw, acc);
    }
    // 5) residual + LayerNorm + ReLU
    {
        int blocks = (N_NODES + 7) / 8;
        finalize_kernel<<<blocks, 256, 0, stream>>>(x, acc, hsk, xw, deg,
                                                    bg, gamma, beta, out);
    }
}